// Model_74440373174855
// MI455X (gfx1250) — compile-verified
//
#include <hip/hip_runtime.h>

typedef __attribute__((ext_vector_type(2))) float v2f;
typedef __attribute__((ext_vector_type(8))) float v8f;

#define NPTS 8192
#define EPSF 1e-5f
#define NCHUNKS 16          // 16 chunks x 512 j
#define JCHUNK 512
#define JTILES (JCHUNK / 16)
#define BLOCK 128           // 4 waves (wave32)
#define MAX_JSPLIT 16

__global__ __launch_bounds__(BLOCK) void nbody_force_wmma(
    const float* __restrict__ pos, float* __restrict__ partial, int jsplit)
{
    __shared__ float4 sj[JCHUNK];   // (x, y, z, r2) per staged j

    const int tid  = threadIdx.x;
    const int lane = tid & 31;
    const int wave = tid >> 5;
    const int n16  = lane & 15;
    const bool hi  = (lane >= 16);

    // This wave's i = ibase + n16 (column N of the WMMA tile)
    const int i = blockIdx.x * 64 + wave * 16 + n16;
    const float xi = pos[i * 3 + 0];
    const float yi = pos[i * 3 + 1];
    const float zi = pos[i * 3 + 2];
    const float ri2 = xi * xi + yi * yi + zi * zi;

    // B fragment (4x16, K x N): col i = (-2xi, -2yi, -2zi, 1)
    // lanes 0-15 hold K=0,1 ; lanes 16-31 hold K=2,3
    v2f b;
    b[0] = hi ? (-2.0f * zi) : (-2.0f * xi);
    b[1] = hi ? 1.0f         : (-2.0f * yi);

    // C fragment: every row of column i preloaded with ri2,
    // so D = A*B + C = rj2 + ri2 - 2*pi.pj = |pi-pj|^2 directly.
    v8f cfrag;
#pragma unroll
    for (int m = 0; m < 8; ++m) cfrag[m] = ri2;

    float fx = 0.0f, fy = 0.0f, fz = 0.0f;

    for (int c = blockIdx.y; c < NCHUNKS; c += jsplit) {
        __syncthreads();
        // Stage 512 j positions (+ r2) into LDS cooperatively
        for (int t = tid; t < JCHUNK; t += BLOCK) {
            const int j = c * JCHUNK + t;
            const float xj = pos[j * 3 + 0];
            const float yj = pos[j * 3 + 1];
            const float zj = pos[j * 3 + 2];
            sj[t] = make_float4(xj, yj, zj, xj * xj + yj * yj + zj * zj);
        }
        __syncthreads();

        for (int jt = 0; jt < JTILES; ++jt) {
            // A fragment (16x4, M x K): row j = (xj, yj, zj, rj2)
            const float4 pj = sj[jt * 16 + n16];
            v2f a;
            a[0] = hi ? pj.z : pj.x;
            a[1] = hi ? pj.w : pj.y;

            // d2[m] = |p_{jb+m} - p_i|^2 for this lane's column i
            v8f d2 = __builtin_amdgcn_wmma_f32_16x16x4_f32(
                false, a, false, b, (short)0, cfrag, false, false);

            const int jb = jt * 16 + (hi ? 8 : 0);
#pragma unroll
            for (int m = 0; m < 8; ++m) {
                const float4 q = sj[jb + m];           // half-wave-uniform LDS broadcast
                float dd = d2[m];
                dd = dd > 0.0f ? dd : 0.0f;            // clamp rounding noise (self-pair)
                const float r = __builtin_amdgcn_sqrtf(dd);
                const float t = r + EPSF;
                const float w = __builtin_amdgcn_rcpf(t * t * t);
                fx = fmaf(w, xi - q.x, fx);            // self-pair: diff==0 exactly -> 0
                fy = fmaf(w, yi - q.y, fy);
                fz = fmaf(w, zi - q.z, fz);
            }
        }
    }

    // lanes L and L+16 accumulated j-halves (M=0..7 vs M=8..15) of the same i
    fx += __shfl_xor(fx, 16, 32);
    fy += __shfl_xor(fy, 16, 32);
    fz += __shfl_xor(fz, 16, 32);

    if (!hi) {
        float* o = partial + ((size_t)blockIdx.y * NPTS + (size_t)i) * 3;
        o[0] = fx; o[1] = fy; o[2] = fz;
    }
}

__global__ void reduce_partials(const float* __restrict__ partial,
                                float* __restrict__ out, int jsplit)
{
    const int idx = blockIdx.x * blockDim.x + threadIdx.x;
    if (idx < NPTS * 3) {
        float s = 0.0f;
        for (int k = 0; k < jsplit; ++k)
            s += partial[(size_t)k * NPTS * 3 + idx];
        out[idx] = s;
    }
}

extern "C" void kernel_launch(void* const* d_in, const int* in_sizes, int n_in,
                              void* d_out, int out_size, void* d_ws, size_t ws_size,
                              hipStream_t stream)
{
    const float* pos = (const float*)d_in[0];
    float* out = (float*)d_out;

    const size_t slice = (size_t)NPTS * 3 * sizeof(float);
    int jsplit = (int)(ws_size / slice);
    if (jsplit > MAX_JSPLIT) jsplit = MAX_JSPLIT;

    float* partial = (float*)d_ws;
    if (jsplit < 1) {            // scratch too small: single split, write d_out directly
        jsplit = 1;
        partial = out;
    }

    dim3 grid(NPTS / 64, jsplit);
    nbody_force_wmma<<<grid, BLOCK, 0, stream>>>(pos, partial, jsplit);

    if (partial != out) {
        const int tot = NPTS * 3;
        reduce_partials<<<(tot + 255) / 256, 256, 0, stream>>>(partial, out, jsplit);
    }
}